// MeshPatchEmbed_9955734192759
// MI455X (gfx1250) — compile-verified
//
#include <hip/hip_runtime.h>
#include <hip/hip_bf16.h>
#include <stdint.h>

// ---------------------------------------------------------------------------
// MeshPatchEmbed for MI455X (gfx1250, wave32)
//  Phase A: voxel binning + per-voxel 32-smallest-area selection (counting
//           sort by voxel + LDS bitonic sort per voxel).
//  Phase B: tokens[32768,288] x W[96,288]^T + b  via V_WMMA_F32_16X16X4_F32.
//           One wave per 16-voxel M-tile carrying all 6 N-tiles (96 embed),
//           so the dominant stream (tokens, 37.7 MB) is read exactly once:
//           ~50 MB total @ 23.3 TB/s ~= 2.2 us, the f32 roofline floor.
// ---------------------------------------------------------------------------

typedef float v2f __attribute__((ext_vector_type(2)));
typedef float v8f __attribute__((ext_vector_type(8)));

#define VGRID 32
#define NVOX  (VGRID * VGRID * VGRID)   // 32768
#define MAXD  32
#define NFEAT 9
#define INC   (MAXD * NFEAT)            // 288
#define EMB   96
#define NT_N  (EMB / 16)                // 6 N-tiles per wave
#define CAP   2048                      // per-voxel candidate cap (mean ~61)

// ---------------- Phase A1: per-face voxel id + area key + histogram -------
__global__ __launch_bounds__(256) void face_prep(
    const float* __restrict__ verts, const int* __restrict__ faces, int F,
    int* __restrict__ lin_arr, unsigned int* __restrict__ areab,
    unsigned int* __restrict__ counts)
{
    int f = blockIdx.x * blockDim.x + threadIdx.x;
    if (f >= F) return;
    int i0 = faces[3 * f + 0], i1 = faces[3 * f + 1], i2 = faces[3 * f + 2];
    float x0 = verts[3 * i0 + 0], y0 = verts[3 * i0 + 1], z0 = verts[3 * i0 + 2];
    float x1 = verts[3 * i1 + 0], y1 = verts[3 * i1 + 1], z1 = verts[3 * i1 + 2];
    float x2 = verts[3 * i2 + 0], y2 = verts[3 * i2 + 1], z2 = verts[3 * i2 + 2];

    const float third = 1.0f / 3.0f;
    float cx = (x0 + x1 + x2) * third;
    float cy = (y0 + y1 + y2) * third;
    float cz = (z0 + z1 + z2) * third;
    int vx = (int)floorf((cx + 1.0f) * 16.0f);
    int vy = (int)floorf((cy + 1.0f) * 16.0f);
    int vz = (int)floorf((cz + 1.0f) * 16.0f);
    vx = vx < 0 ? 0 : (vx > VGRID - 1 ? VGRID - 1 : vx);
    vy = vy < 0 ? 0 : (vy > VGRID - 1 ? VGRID - 1 : vy);
    vz = vz < 0 ? 0 : (vz > VGRID - 1 ? VGRID - 1 : vz);
    int lin = (vx * VGRID + vy) * VGRID + vz;

    float abx = x1 - x0, aby = y1 - y0, abz = z1 - z0;
    float acx = x2 - x0, acy = y2 - y0, acz = z2 - z0;
    float crx = aby * acz - abz * acy;
    float cry = abz * acx - abx * acz;
    float crz = abx * acy - aby * acx;
    float area = crx * crx + cry * cry + crz * crz;   // >= 0: bit pattern is order-preserving

    lin_arr[f] = lin;
    areab[f]   = __float_as_uint(area);
    atomicAdd(&counts[lin], 1u);
}

// ---------------- Phase A2: exclusive scan of 32768 counts (one block) -----
__global__ __launch_bounds__(1024) void scan_offsets(
    const unsigned int* __restrict__ counts,
    unsigned int* __restrict__ offsets,
    unsigned int* __restrict__ cursors)
{
    __shared__ unsigned int sums[1024];
    const int t = threadIdx.x;
    const int CHUNK = NVOX / 1024;          // 32
    const int base = t * CHUNK;
    unsigned int s = 0;
    #pragma unroll
    for (int i = 0; i < CHUNK; ++i) s += counts[base + i];
    sums[t] = s;
    __syncthreads();
    // Hillis-Steele inclusive scan (read phase / write phase separated)
    for (int off = 1; off < 1024; off <<= 1) {
        unsigned int v = (t >= off) ? sums[t - off] : 0u;
        __syncthreads();
        sums[t] += v;
        __syncthreads();
    }
    unsigned int run = sums[t] - s;         // exclusive base for this chunk
    #pragma unroll
    for (int i = 0; i < CHUNK; ++i) {
        offsets[base + i] = run;
        cursors[base + i] = 0u;
        run += counts[base + i];
    }
}

// ---------------- Phase A3: scatter 64-bit keys into voxel buckets ---------
__global__ __launch_bounds__(256) void scatter_keys(
    const int* __restrict__ lin_arr, const unsigned int* __restrict__ areab,
    const unsigned int* __restrict__ offsets, unsigned int* __restrict__ cursors,
    unsigned long long* __restrict__ keys, int F)
{
    int f = blockIdx.x * blockDim.x + threadIdx.x;
    if (f >= F) return;
    int lin = lin_arr[f];
    unsigned int pos = offsets[lin] + atomicAdd(&cursors[lin], 1u);
    keys[pos] = ((unsigned long long)areab[f] << 32) | (unsigned int)f;
}

// ---------------- Phase A4: per-voxel bitonic sort + token write -----------
__global__ __launch_bounds__(256) void voxel_select(
    const unsigned long long* __restrict__ keys,
    const unsigned int* __restrict__ counts,
    const unsigned int* __restrict__ offsets,
    const float* __restrict__ verts, const int* __restrict__ faces,
    float* __restrict__ tokens)
{
    __shared__ unsigned long long sk[CAP];
    const int v = blockIdx.x;
    unsigned int n      = counts[v];
    unsigned int off    = offsets[v];
    unsigned int capped = n > CAP ? CAP : n;
    unsigned int P = 32;
    while (P < capped) P <<= 1;             // power-of-two size, <= CAP

    for (unsigned int i = threadIdx.x; i < P; i += blockDim.x)
        sk[i] = (i < capped) ? keys[off + i] : ~0ull;
    __syncthreads();

    for (unsigned int k = 2; k <= P; k <<= 1) {
        for (unsigned int j = k >> 1; j > 0; j >>= 1) {
            for (unsigned int i = threadIdx.x; i < P; i += blockDim.x) {
                unsigned int ixj = i ^ j;
                if (ixj > i) {
                    unsigned long long a = sk[i], b2 = sk[ixj];
                    bool up = ((i & k) == 0);
                    if ((a > b2) == up) { sk[i] = b2; sk[ixj] = a; }
                }
            }
            __syncthreads();
        }
    }

    unsigned int m = n < MAXD ? n : MAXD;
    for (unsigned int r = threadIdx.x; r < m; r += blockDim.x) {
        int f  = (int)(unsigned int)(sk[r] & 0xffffffffull);
        int i0 = faces[3 * f + 0], i1 = faces[3 * f + 1], i2 = faces[3 * f + 2];
        float* o = tokens + (size_t)v * INC + r * NFEAT;
        o[0] = verts[3 * i0 + 0]; o[1] = verts[3 * i0 + 1]; o[2] = verts[3 * i0 + 2];
        o[3] = verts[3 * i1 + 0]; o[4] = verts[3 * i1 + 1]; o[5] = verts[3 * i1 + 2];
        o[6] = verts[3 * i2 + 0]; o[7] = verts[3 * i2 + 1]; o[8] = verts[3 * i2 + 2];
    }
}

// ---------------- Phase B: f32 WMMA GEMM  out[e,v] = W.tokens^T + b --------
// One wave32 per 16-voxel M-tile; 6 accumulators cover all 96 embed channels,
// so each A (tokens) load feeds 6 WMMAs and tokens is streamed exactly once.
__global__ __launch_bounds__(256) void gemm_wmma(
    const float* __restrict__ tokens,   // [NVOX, 288]
    const float* __restrict__ Wm,       // [96, 288]
    const float* __restrict__ bias,     // [96]
    float* __restrict__ out)            // [96, NVOX]
{
    int wave  = (blockIdx.x * blockDim.x + threadIdx.x) >> 5;    // 2048 waves
    int lane  = threadIdx.x & 31;
    int hf    = lane >> 4;                                       // half-wave
    int idx   = lane & 15;
    int mTile = wave;

    // A 16x4 f32 layout: lanes0-15 M=0..15 K={0,1}; lanes16-31 M=0..15 K={2,3}
    const float* aPtr = tokens + (size_t)(mTile * 16 + idx) * INC + 2 * hf;
    // B 4x16 f32 layout mirrors A with N on lanes (B[k,n] = W[n,k]);
    // N-tile t adds 16 rows of W.
    const float* bPtr = Wm + (size_t)idx * INC + 2 * hf;

    v8f acc[NT_N];
    #pragma unroll
    for (int t = 0; t < NT_N; ++t) {
        float bv = bias[t * 16 + idx];                           // bias depends on N only
        #pragma unroll
        for (int r = 0; r < 8; ++r) acc[t][r] = bv;
    }

    #pragma unroll 2
    for (int k = 0; k < INC; k += 4) {
        v2f a = *(const v2f*)(aPtr + k);
        #pragma unroll
        for (int t = 0; t < NT_N; ++t) {
            v2f b = *(const v2f*)(bPtr + (size_t)t * 16 * INC + k);
            acc[t] = __builtin_amdgcn_wmma_f32_16x16x4_f32(
                /*neg_a=*/false, a, /*neg_b=*/false, b,
                /*c_mod=*/(short)0, acc[t], /*reuse_a=*/false, /*reuse_b=*/false);
        }
    }

    // D layout: VGPR r -> D[m = r + 8*hf][n = idx]
    #pragma unroll
    for (int t = 0; t < NT_N; ++t) {
        float* o = out + (size_t)(t * 16 + idx) * NVOX + (size_t)(mTile * 16 + 8 * hf);
        #pragma unroll
        for (int r = 0; r < 8; ++r) o[r] = acc[t][r];
    }
}

// ---------------------------------------------------------------------------
extern "C" void kernel_launch(void* const* d_in, const int* in_sizes, int n_in,
                              void* d_out, int out_size, void* d_ws, size_t ws_size,
                              hipStream_t stream) {
    const float* verts = (const float*)d_in[0];
    const int*   faces = (const int*)d_in[1];
    const float* Wm    = (const float*)d_in[2];
    const float* bias  = (const float*)d_in[3];
    float*       out   = (float*)d_out;
    const int F = in_sizes[1] / 3;          // 2,000,000 faces

    // Workspace carve-up (~70.2 MB total for F = 2M):
    char* ws = (char*)d_ws;
    float* tokens = (float*)ws;
    size_t tokBytes = (size_t)NVOX * INC * sizeof(float);        // 37,748,736 B
    unsigned int* counts  = (unsigned int*)(ws + tokBytes);
    unsigned int* offsets = counts + NVOX;
    unsigned int* cursors = offsets + NVOX;
    int*          lin_arr = (int*)(cursors + NVOX);
    unsigned int* areab   = (unsigned int*)(lin_arr + F);
    uintptr_t kp = (uintptr_t)(areab + F);
    kp = (kp + 7) & ~(uintptr_t)7;
    unsigned long long* keys = (unsigned long long*)kp;

    hipMemsetAsync(tokens, 0, tokBytes, stream);                 // zero-padded tokens
    hipMemsetAsync(counts, 0, NVOX * sizeof(unsigned int), stream);

    int fb = (F + 255) / 256;
    face_prep   <<<fb,    256,  0, stream>>>(verts, faces, F, lin_arr, areab, counts);
    scan_offsets<<<1,     1024, 0, stream>>>(counts, offsets, cursors);
    scatter_keys<<<fb,    256,  0, stream>>>(lin_arr, areab, offsets, cursors, keys, F);
    voxel_select<<<NVOX,  256,  0, stream>>>(keys, counts, offsets, verts, faces, tokens);

    // NVOX/16 = 2048 waves, 8 waves (256 thr) per block -> 256 blocks
    gemm_wmma   <<<256,   256,  0, stream>>>(tokens, Wm, bias, out);
}